// Cheby_78795470012808
// MI455X (gfx1250) — compile-verified
//
#include <hip/hip_runtime.h>
#include <hip/hip_bf16.h>

typedef __attribute__((ext_vector_type(2))) float v2f;
typedef __attribute__((ext_vector_type(8))) float v8f;

// ---------------------------------------------------------------------------
// Utility kernels
// ---------------------------------------------------------------------------
__global__ __launch_bounds__(256) void zero_f32_kernel(float* __restrict__ p, int n4) {
    int i = blockIdx.x * blockDim.x + threadIdx.x;
    if (i < n4) ((float4*)p)[i] = make_float4(0.f, 0.f, 0.f, 0.f);
}

__global__ __launch_bounds__(256) void negcopy_kernel(float* __restrict__ dst,
                                                      const float* __restrict__ src, int n4) {
    int i = blockIdx.x * blockDim.x + threadIdx.x;
    if (i < n4) {
        float4 v = ((const float4*)src)[i];
        ((float4*)dst)[i] = make_float4(-v.x, -v.y, -v.z, -v.w);
    }
}

// Pack W[3][64][NOUT] -> Wp[3][32][NOUT][2]: adjacent K rows interleaved per
// column so the WMMA B operand (k, k+1) is one aligned b64 load.
__global__ __launch_bounds__(256) void packw_kernel(const float* __restrict__ W,
                                                    float* __restrict__ Wp,
                                                    int nout, int total) {
    int idx = blockIdx.x * blockDim.x + threadIdx.x;
    if (idx >= total) return;                 // total = 3*64*nout
    int col = idx % nout;
    int k   = (idx / nout) & 63;
    int m   = idx / (nout * 64);
    Wp[(((size_t)m * 32 + (k >> 1)) * nout + col) * 2 + (k & 1)] = W[idx];
}

// ---------------------------------------------------------------------------
// Graph normalization
// ---------------------------------------------------------------------------
__global__ __launch_bounds__(256) void deg_kernel(const int* __restrict__ src,
                                                  const int* __restrict__ dst,
                                                  float* __restrict__ deg, int nedges) {
    int e = blockIdx.x * blockDim.x + threadIdx.x;
    if (e >= nedges) return;
    int s = src[e];
    if (s != dst[e]) unsafeAtomicAdd(deg + s, 1.0f);
}

__global__ __launch_bounds__(256) void dis_kernel(float* __restrict__ deg, int n) {
    int i = blockIdx.x * blockDim.x + threadIdx.x;
    if (i >= n) return;
    float d = deg[i];
    deg[i] = (d > 0.f) ? rsqrtf(fmaxf(d, 1.0f)) : 0.f;
}

__global__ __launch_bounds__(256) void edgew_kernel(const int* __restrict__ src,
                                                    const int* __restrict__ dst,
                                                    const float* __restrict__ dis,
                                                    float* __restrict__ w, int nedges) {
    int e = blockIdx.x * blockDim.x + threadIdx.x;
    if (e >= nedges) return;
    int s = src[e], d = dst[e];
    w[e] = (s != d) ? (-dis[s] * dis[d]) : 0.f;
}

// ---------------------------------------------------------------------------
// SpMM: out[dst] += scale * w[e] * x[src]   (64 features, 16 threads/edge x f32x4)
// ---------------------------------------------------------------------------
__global__ __launch_bounds__(256) void spmm_kernel(const int* __restrict__ src,
                                                   const int* __restrict__ dst,
                                                   const float* __restrict__ w,
                                                   const float* __restrict__ x,
                                                   float* __restrict__ out,
                                                   int nedges, float scale) {
    long long tid = (long long)blockIdx.x * blockDim.x + threadIdx.x;
    int e = (int)(tid >> 4);
    if (e >= nedges) return;
    float we = w[e] * scale;
    if (we == 0.f) return;
    int fg = (int)(tid & 15);
    int s = src[e], d = dst[e];
    float4 xv = ((const float4*)x)[s * 16 + fg];
    float* o = out + (size_t)d * 64 + fg * 4;
    unsafeAtomicAdd(o + 0, we * xv.x);
    unsafeAtomicAdd(o + 1, we * xv.y);
    unsafeAtomicAdd(o + 2, we * xv.z);
    unsafeAtomicAdd(o + 3, we * xv.w);
}

// ---------------------------------------------------------------------------
// Fused Chebyshev GEMM: out = A0*W[0] + A1*W[1] + A2*W[2] + bias (opt. ReLU)
// One wave handles TWO 16-row strips (32 rows) so each B b64 load feeds two
// V_WMMA_F32_16X16X4_F32 issues.
// ---------------------------------------------------------------------------
template <int NT, int NOUT>
__device__ inline void cheb_accum2(const float* __restrict__ A,
                                   const float* __restrict__ Wp,  // [32][NOUT][2]
                                   int rowA0, int rowA1, int khalf, int colLane,
                                   v8f (&acc0)[NT], v8f (&acc1)[NT]) {
    const float* ar0 = A + (size_t)rowA0 * 64;
    const float* ar1 = A + (size_t)rowA1 * 64;
    #pragma unroll
    for (int k0 = 0; k0 < 64; k0 += 4) {
        int kk = k0 + 2 * khalf;               // lanes 16-31 carry K offset +2
        v2f a0 = *(const v2f*)(ar0 + kk);      // kk even -> 8B aligned
        v2f a1 = *(const v2f*)(ar1 + kk);
        int p = (k0 >> 1) + khalf;             // packed pair index = kk/2
        #pragma unroll
        for (int t = 0; t < NT; ++t) {
            int cg = t * 16 + colLane;
            v2f b;
            if ((NOUT & 15) == 0 || cg < NOUT) {
                b = *(const v2f*)(Wp + ((size_t)p * NOUT + cg) * 2);
            } else {
                b.x = 0.f;
                b.y = 0.f;
            }
            acc0[t] = __builtin_amdgcn_wmma_f32_16x16x4_f32(
                false, a0, false, b, (short)0, acc0[t], false, false);
            acc1[t] = __builtin_amdgcn_wmma_f32_16x16x4_f32(
                false, a1, false, b, (short)0, acc1[t], false, false);
        }
    }
}

template <int NOUT, bool RELU>
__global__ __launch_bounds__(256) void cheb_gemm_kernel(const float* __restrict__ A0,
                                                        const float* __restrict__ A1,
                                                        const float* __restrict__ A2,
                                                        const float* __restrict__ Wp,
                                                        const float* __restrict__ bias,
                                                        float* __restrict__ out, int nodes) {
    constexpr int NT  = (NOUT + 15) / 16;
    constexpr int PSZ = 64 * NOUT;             // floats per packed W matrix
    int wave = (blockIdx.x * blockDim.x + threadIdx.x) >> 5;
    int lane = threadIdx.x & 31;
    int l15  = lane & 15;                      // A row-in-strip / B,C,D column
    int khalf = lane >> 4;                     // K half (A/B), M half (C/D)
    int row0 = wave * 32;
    if (row0 >= nodes) return;

    v8f acc0[NT] = {};
    v8f acc1[NT] = {};
    cheb_accum2<NT, NOUT>(A0, Wp + 0 * PSZ, row0 + l15, row0 + 16 + l15, khalf, l15, acc0, acc1);
    cheb_accum2<NT, NOUT>(A1, Wp + 1 * PSZ, row0 + l15, row0 + 16 + l15, khalf, l15, acc0, acc1);
    cheb_accum2<NT, NOUT>(A2, Wp + 2 * PSZ, row0 + l15, row0 + 16 + l15, khalf, l15, acc0, acc1);

    #pragma unroll
    for (int t = 0; t < NT; ++t) {
        int col = t * 16 + l15;
        if ((NOUT & 15) != 0 && col >= NOUT) continue;
        float bv = bias[col];
        #pragma unroll
        for (int i = 0; i < 8; ++i) {
            int r = row0 + i + 8 * khalf;      // C/D: VGPR i -> M=i (+8 for hi lanes)
            float v0 = acc0[t][i] + bv;
            float v1 = acc1[t][i] + bv;
            if (RELU) { v0 = fmaxf(v0, 0.f); v1 = fmaxf(v1, 0.f); }
            if (r < nodes)      out[(size_t)r * NOUT + col]        = v0;
            if (r + 16 < nodes) out[(size_t)(r + 16) * NOUT + col] = v1;
        }
    }
}

// ---------------------------------------------------------------------------
// Row softmax over 40 classes, in place
// ---------------------------------------------------------------------------
__global__ __launch_bounds__(256) void softmax40_kernel(float* __restrict__ out, int nodes) {
    int i = blockIdx.x * blockDim.x + threadIdx.x;
    if (i >= nodes) return;
    float* row = out + (size_t)i * 40;
    float m = -INFINITY;
    #pragma unroll
    for (int c = 0; c < 40; ++c) m = fmaxf(m, row[c]);
    float e[40];
    float s = 0.f;
    #pragma unroll
    for (int c = 0; c < 40; ++c) { e[c] = __expf(row[c] - m); s += e[c]; }
    float inv = 1.f / s;
    #pragma unroll
    for (int c = 0; c < 40; ++c) row[c] = e[c] * inv;
}

// ---------------------------------------------------------------------------
// Host launcher
// ---------------------------------------------------------------------------
extern "C" void kernel_launch(void* const* d_in, const int* in_sizes, int n_in,
                              void* d_out, int out_size, void* d_ws, size_t ws_size,
                              hipStream_t stream) {
    const float* x  = (const float*)d_in[0];
    const int*   ei = (const int*)d_in[1];
    const float* W1 = (const float*)d_in[2];
    const float* b1 = (const float*)d_in[3];
    const float* W2 = (const float*)d_in[4];
    const float* b2 = (const float*)d_in[5];
    float* out = (float*)d_out;

    const int N = in_sizes[0] / 64;       // 100000 nodes
    const int E = in_sizes[1] / 2;        // 1600000 edges
    const int* src = ei;
    const int* dst = ei + E;

    // Workspace carve-up (256B aligned)
    auto align256 = [](size_t v) { return (v + 255) & ~(size_t)255; };
    char* ws = (char*)d_ws;
    size_t off = 0;
    float* dis  = (float*)(ws + off); off += align256((size_t)N * 4);            // deg -> dis
    float* wbuf = (float*)(ws + off); off += align256((size_t)E * 4);            // edge weights
    float* A1   = (float*)(ws + off); off += align256((size_t)N * 64 * 4);       // Tx1
    float* A2   = (float*)(ws + off); off += align256((size_t)N * 64 * 4);       // Tx2
    float* H    = (float*)(ws + off); off += align256((size_t)N * 64 * 4);       // hidden
    float* Wp1  = (float*)(ws + off); off += align256((size_t)3 * 64 * 64 * 4);  // packed W1
    float* Wp2  = (float*)(ws + off); off += align256((size_t)3 * 64 * 40 * 4);  // packed W2
    (void)ws_size; (void)n_in; (void)out_size;

    const int TB = 256;
    const int nodeBlk  = (N + TB - 1) / TB;
    const int edgeBlk  = (E + TB - 1) / TB;
    const int feat4Blk = ((N * 64 / 4) + TB - 1) / TB;
    const long long spmmThreads = (long long)E * 16;
    const int spmmBlk  = (int)((spmmThreads + TB - 1) / TB);
    const int gemmBlk  = ((((N + 31) / 32) * 32) + TB - 1) / TB;   // 1 wave / 32 rows

    // --- weight packing + normalization ---
    packw_kernel<<<(3 * 64 * 64 + TB - 1) / TB, TB, 0, stream>>>(W1, Wp1, 64, 3 * 64 * 64);
    packw_kernel<<<(3 * 64 * 40 + TB - 1) / TB, TB, 0, stream>>>(W2, Wp2, 40, 3 * 64 * 40);
    zero_f32_kernel<<<(N / 4 + TB - 1) / TB, TB, 0, stream>>>(dis, N / 4);
    deg_kernel<<<edgeBlk, TB, 0, stream>>>(src, dst, dis, E);
    dis_kernel<<<nodeBlk, TB, 0, stream>>>(dis, N);
    edgew_kernel<<<edgeBlk, TB, 0, stream>>>(src, dst, dis, wbuf, E);

    // --- layer 1 ---
    zero_f32_kernel<<<feat4Blk, TB, 0, stream>>>(A1, N * 64 / 4);
    spmm_kernel<<<spmmBlk, TB, 0, stream>>>(src, dst, wbuf, x, A1, E, 1.0f);    // Tx1
    negcopy_kernel<<<feat4Blk, TB, 0, stream>>>(A2, x, N * 64 / 4);             // -Tx0
    spmm_kernel<<<spmmBlk, TB, 0, stream>>>(src, dst, wbuf, A1, A2, E, 2.0f);   // Tx2
    cheb_gemm_kernel<64, true><<<gemmBlk, TB, 0, stream>>>(x, A1, A2, Wp1, b1, H, N);

    // --- layer 2 ---
    zero_f32_kernel<<<feat4Blk, TB, 0, stream>>>(A1, N * 64 / 4);
    spmm_kernel<<<spmmBlk, TB, 0, stream>>>(src, dst, wbuf, H, A1, E, 1.0f);    // Tx1'
    negcopy_kernel<<<feat4Blk, TB, 0, stream>>>(A2, H, N * 64 / 4);             // -Tx0'
    spmm_kernel<<<spmmBlk, TB, 0, stream>>>(src, dst, wbuf, A1, A2, E, 2.0f);   // Tx2'
    cheb_gemm_kernel<40, false><<<gemmBlk, TB, 0, stream>>>(H, A1, A2, Wp2, b2, out, N);

    // --- softmax ---
    softmax40_kernel<<<nodeBlk, TB, 0, stream>>>(out, N);
}